// BM3_52304111730855
// MI455X (gfx1250) — compile-verified
//
#include <hip/hip_runtime.h>

typedef float v2f __attribute__((ext_vector_type(2)));
typedef float v8f __attribute__((ext_vector_type(8)));

#define N_USERS 50000
#define N_ITEMS 30000
#define N_NODES 80000
#define DIM     128
#define NLAYER  2

// ---------------- degree / norm ----------------

__global__ __launch_bounds__(256) void k_init_deg(float* deg) {
  int n = blockIdx.x * 256 + threadIdx.x;
  if (n < N_NODES) deg[n] = 1.0f;            // self-loop contributes 1
}

__global__ __launch_bounds__(256) void k_count_deg(const int* __restrict__ col, int E,
                                                   float* __restrict__ deg) {
  int e = blockIdx.x * 256 + threadIdx.x;
  if (e < E) unsafeAtomicAdd(&deg[col[e]], 1.0f);   // -> global_atomic_add_f32 (no return)
}

__global__ __launch_bounds__(256) void k_dinv(float* deg) {
  int n = blockIdx.x * 256 + threadIdx.x;
  if (n < N_NODES) deg[n] = rsqrtf(deg[n]);  // deg >= 1 always (self-loops)
}

// ---------------- ego init (also seeds the accumulator in d_out) ----------------

__global__ __launch_bounds__(256) void k_init_ego(const float4* __restrict__ user,
                                                  const float4* __restrict__ item,
                                                  float4* __restrict__ ego,
                                                  float4* __restrict__ acc) {
  int i = blockIdx.x * 256 + threadIdx.x;                 // float4 index
  if (i >= N_NODES * DIM / 4) return;
  float4 v = (i < N_USERS * DIM / 4) ? user[i] : item[i - N_USERS * DIM / 4];
  ego[i] = v;
  acc[i] = v;
}

// ---------------- GEMM: y = x @ W^T  via V_WMMA_F32_16X16X4_F32 ----------------
// One wave computes one 16x16 tile of y; K=128 chained as 32 WMMA(K=4) steps.
// A (16x4 f32): lane<16 holds M=lane; VGPR v holds K = 2*half + v.
// B (4x16 f32): mirrors A; B[kk][j] = W[j0+j][k0+kk] (row-major W, x@W^T).
// C/D (16x16 f32, 8 VGPRs): VGPR v -> M = v + 8*half, N = lane&15.
__global__ __launch_bounds__(256) void k_gemm_xwT(const float* __restrict__ x,
                                                  const float* __restrict__ W,
                                                  float* __restrict__ y) {
  const int warp = threadIdx.x >> 5;
  const int lane = threadIdx.x & 31;
  const int half = lane >> 4;
  const int mr   = lane & 15;
  const int n0   = blockIdx.x * 16;   // row tile  (80000/16 = 5000 blocks)
  const int j0   = warp * 16;         // col tile  (8 warps cover 128 cols)

  const float* xa = x + (size_t)(n0 + mr) * DIM + 2 * half;
  const float* wb = W + (size_t)(j0 + mr) * DIM + 2 * half;

  v8f c = {0.f, 0.f, 0.f, 0.f, 0.f, 0.f, 0.f, 0.f};
#pragma unroll
  for (int k = 0; k < DIM / 4; ++k) {
    v2f a = *(const v2f*)(xa + 4 * k);   // x[n0+mr][4k + 2*half + {0,1}]
    v2f b = *(const v2f*)(wb + 4 * k);   // W[j0+mr][4k + 2*half + {0,1}]
    c = __builtin_amdgcn_wmma_f32_16x16x4_f32(false, a, false, b, (short)0, c,
                                              false, false);
  }

  float* yo = y + (size_t)(n0 + 8 * half) * DIM + j0 + mr;
#pragma unroll
  for (int v = 0; v < 8; ++v) yo[(size_t)v * DIM] = c[v];
}

// ---------------- per-layer: dst = bias + self-loop message (non-atomic init) ----------------

__global__ __launch_bounds__(256) void k_selfloop_bias(const float4* __restrict__ xw,
                                                       const float* __restrict__ dinv,
                                                       const float* __restrict__ bias,
                                                       float4* __restrict__ dst) {
  int i = blockIdx.x * 256 + threadIdx.x;                 // float4 index
  if (i >= N_NODES * DIM / 4) return;
  int n  = (i * 4) >> 7;                                  // node id (128 | 4)
  int d4 = i & (DIM / 4 - 1);
  float s = dinv[n]; s = s * s;                           // norm of self loop
  const float4* b4 = (const float4*)bias;
  float4 xv = xw[i], bv = b4[d4], o;
  o.x = bv.x + xv.x * s;
  o.y = bv.y + xv.y * s;
  o.z = bv.z + xv.z * s;
  o.w = bv.w + xv.w * s;
  dst[i] = o;
}

// ---------------- edge scatter: dst[col] += xw[row] * dinv[row]*dinv[col] ----------------
// wave32-native: one wave per edge, each lane does 4 strided f32 atomics.
__global__ __launch_bounds__(256) void k_scatter(const int* __restrict__ row,
                                                 const int* __restrict__ col,
                                                 const float* __restrict__ dinv,
                                                 const float* __restrict__ src,
                                                 float* __restrict__ dst, int E) {
  int warp = threadIdx.x >> 5;
  int lane = threadIdx.x & 31;
  int e = blockIdx.x * 8 + warp;
  if (e >= E) return;
  int r = row[e], c = col[e];
  float nrm = dinv[r] * dinv[c];
  const float* s = src + (size_t)r * DIM;
  float* d       = dst + (size_t)c * DIM;
#pragma unroll
  for (int i = 0; i < 4; ++i) {
    int dd = lane + 32 * i;
    unsafeAtomicAdd(&d[dd], s[dd] * nrm);
  }
}

// ---------------- acc += layer output ----------------

__global__ __launch_bounds__(256) void k_acc(const float4* __restrict__ a,
                                             float4* __restrict__ out) {
  int i = blockIdx.x * 256 + threadIdx.x;
  if (i >= N_NODES * DIM / 4) return;
  float4 o = out[i], v = a[i];
  o.x += v.x; o.y += v.y; o.z += v.z; o.w += v.w;
  out[i] = o;
}

// ---------------- finalize: /3, item part gets +item_emb ----------------

__global__ __launch_bounds__(256) void k_final(const float4* __restrict__ item,
                                               float4* __restrict__ out) {
  int i = blockIdx.x * 256 + threadIdx.x;
  if (i >= N_NODES * DIM / 4) return;
  const float inv3 = 1.0f / 3.0f;
  float4 o = out[i];
  o.x *= inv3; o.y *= inv3; o.z *= inv3; o.w *= inv3;
  if (i >= N_USERS * DIM / 4) {
    float4 t = item[i - N_USERS * DIM / 4];
    o.x += t.x; o.y += t.y; o.z += t.z; o.w += t.w;
  }
  out[i] = o;
}

// ---------------- driver ----------------

extern "C" void kernel_launch(void* const* d_in, const int* in_sizes, int n_in,
                              void* d_out, int out_size, void* d_ws, size_t ws_size,
                              hipStream_t stream) {
  const float* user = (const float*)d_in[0];   // (50000,128) f32
  const float* item = (const float*)d_in[1];   // (30000,128) f32
  const float* gcnW = (const float*)d_in[2];   // (2,128,128) f32
  const float* gcnB = (const float*)d_in[3];   // (2,128)     f32
  const int*   ei   = (const int*)d_in[4];     // (2,E)       i32
  const int E = in_sizes[4] / 2;               // 2,000,000

  float* out  = (float*)d_out;                 // acc lives here
  float* ws   = (float*)d_ws;
  float* dinv = ws;                            // N floats (deg -> dinv in place)
  float* ego  = ws + N_NODES;                  // N*D floats (320000 B offset, 16B aligned)
  float* xw   = ego + (size_t)N_NODES * DIM;   // N*D floats

  const int* rowp = ei;
  const int* colp = ei + E;

  const dim3 blk(256);
  const int nd4 = N_NODES * DIM / 4;           // float4 elements

  k_init_deg <<<(N_NODES + 255) / 256, blk, 0, stream>>>(dinv);
  k_count_deg<<<(E + 255) / 256,       blk, 0, stream>>>(colp, E, dinv);
  k_dinv     <<<(N_NODES + 255) / 256, blk, 0, stream>>>(dinv);
  k_init_ego <<<(nd4 + 255) / 256,     blk, 0, stream>>>((const float4*)user,
                                                         (const float4*)item,
                                                         (float4*)ego, (float4*)out);
  for (int l = 0; l < NLAYER; ++l) {
    k_gemm_xwT<<<N_NODES / 16, blk, 0, stream>>>(ego, gcnW + (size_t)l * DIM * DIM, xw);
    // scatter destination reuses the (now dead) ego buffer; seed with bias + self-loop msg
    k_selfloop_bias<<<(nd4 + 255) / 256, blk, 0, stream>>>((const float4*)xw, dinv,
                                                           gcnB + (size_t)l * DIM,
                                                           (float4*)ego);
    k_scatter<<<(E + 7) / 8, blk, 0, stream>>>(rowp, colp, dinv, xw, ego, E);
    k_acc    <<<(nd4 + 255) / 256, blk, 0, stream>>>((const float4*)ego, (float4*)out);
  }
  k_final<<<(nd4 + 255) / 256, blk, 0, stream>>>((const float4*)item, (float4*)out);
}